// WRGCN_28243704938828
// MI455X (gfx1250) — compile-verified
//
#include <hip/hip_runtime.h>
#include <hip/hip_bf16.h>

typedef __attribute__((ext_vector_type(2))) float v2f;
typedef __attribute__((ext_vector_type(8))) float v8f;

#define RG_NODES 50000
#define RG_EDGES 600000
#define RG_DIM   128

// Packed weight layout per matrix: [8 n-tiles][32 k-steps][32 lanes] float2
//   entry = { W[kb][nt*16 + (lane&15)], W[kb+1][...] },  kb = 4*k + ((lane>>4)<<1)
// -> in the GEMM inner loop each lane does one coalesced global_load_b64 at an
//    immediate offset (k*256) from a per-n-tile base. 10 matrices total (2 layers
//    x {3 rel, root, skip}) = 0.64 MB, permanently L2-hot.
#define PACK_M 8192   // float2 per matrix (8*32*32)

__global__ __launch_bounds__(256)
void rgcn_pack_w(const float* __restrict__ w_rel,   // [2,3,128,128]
                 const float* __restrict__ w_root,  // [2,128,128]
                 const float* __restrict__ w_skip,  // [2,128,128]
                 float2* __restrict__ pack)         // [2][5][PACK_M]
{
    const int t = blockIdx.x * 256 + threadIdx.x;
    if (t >= 10 * PACK_M) return;
    const int lane = t & 31;
    const int k    = (t >> 5) & 31;
    const int nt   = (t >> 10) & 7;
    const int m    = t >> 13;         // 0..9
    const int l    = m / 5;
    const int mat  = m % 5;
    const float* W = (mat < 3)  ? (w_rel  + ((size_t)l * 3 + mat) * RG_DIM * RG_DIM)
                   : (mat == 3) ? (w_root + (size_t)l * RG_DIM * RG_DIM)
                                : (w_skip + (size_t)l * RG_DIM * RG_DIM);
    const int kb  = 4 * k + ((lane >> 4) << 1);
    const int col = nt * 16 + (lane & 15);
    pack[t] = make_float2(W[kb * RG_DIM + col], W[(kb + 1) * RG_DIM + col]);
}

// ---------------------------------------------------------------------------
// Fused 5-GEMM kernel, one layer. grid = ceil(N/32), block = 128 (4 waves).
// Each block stages 32x128 h/x tiles in LDS; each wave owns one weight matrix
// and computes TWO 16-row A tiles per B load (2 wmma per global_load_b64).
//   waves 0..2: hr[r] = h @ w_rel[r];   wave 3: base = h@w_root + x@w_skip + b
// ---------------------------------------------------------------------------
__global__ __launch_bounds__(128)
void rgcn_gemm_kernel(const float*  __restrict__ h,      // [N,128] layer input
                      const float*  __restrict__ x,      // [N,128] skip input
                      const float2* __restrict__ wp,     // [5][PACK_M] packed (this layer)
                      const float*  __restrict__ b_conv, // [128]
                      const float*  __restrict__ b_skip, // [128]
                      float* __restrict__ hr,            // [3,N,128]
                      float* __restrict__ base,          // [N,128]
                      int n_nodes)
{
    __shared__ float sh[32 * RG_DIM];
    __shared__ float sx[32 * RG_DIM];

    const int tid  = threadIdx.x;
    const int row0 = blockIdx.x * 32;

    for (int i = tid; i < 32 * 32; i += 128) {
        const int r  = i >> 5;
        const int c4 = i & 31;
        const int gr = row0 + r;
        float4 vh = make_float4(0.f, 0.f, 0.f, 0.f);
        float4 vx = vh;
        if (gr < n_nodes) {
            vh = ((const float4*)(h + (size_t)gr * RG_DIM))[c4];
            vx = ((const float4*)(x + (size_t)gr * RG_DIM))[c4];
        }
        ((float4*)(sh + r * RG_DIM))[c4] = vh;
        ((float4*)(sx + r * RG_DIM))[c4] = vx;
    }
    __syncthreads();

    const int wave  = tid >> 5;           // uniform within wave
    const int lane  = tid & 31;
    const int mrow  = lane & 15;
    const int khalf = (lane >> 4) << 1;   // 0 or 2

    const v2f* Bp  = (const v2f*)(wp + (size_t)((wave < 3) ? wave : 3) * PACK_M);
    float*     out = (wave < 3) ? (hr + (size_t)wave * n_nodes * RG_DIM) : base;

    for (int nt = 0; nt < 8; ++nt) {
        v8f acc0 = {};
        v8f acc1 = {};
        const v2f* Bnt = Bp + nt * 1024;

#pragma unroll
        for (int k = 0; k < 32; ++k) {
            const int kb = k * 4 + khalf;
            const v2f b  = Bnt[k * 32 + lane];            // global_load_b64, imm offset
            v2f a0, a1;
            a0.x = sh[mrow * RG_DIM + kb];                // ds_load_b64
            a0.y = sh[mrow * RG_DIM + kb + 1];
            a1.x = sh[(mrow + 16) * RG_DIM + kb];
            a1.y = sh[(mrow + 16) * RG_DIM + kb + 1];
            acc0 = __builtin_amdgcn_wmma_f32_16x16x4_f32(
                false, a0, false, b, (short)0, acc0, false, false);
            acc1 = __builtin_amdgcn_wmma_f32_16x16x4_f32(
                false, a1, false, b, (short)0, acc1, false, false);
        }

        if (wave == 3) {                                  // + x @ w_skip + biases
            const v2f* Bs = (const v2f*)(wp + (size_t)4 * PACK_M) + nt * 1024;
#pragma unroll
            for (int k = 0; k < 32; ++k) {
                const int kb = k * 4 + khalf;
                const v2f b  = Bs[k * 32 + lane];
                v2f a0, a1;
                a0.x = sx[mrow * RG_DIM + kb];
                a0.y = sx[mrow * RG_DIM + kb + 1];
                a1.x = sx[(mrow + 16) * RG_DIM + kb];
                a1.y = sx[(mrow + 16) * RG_DIM + kb + 1];
                acc0 = __builtin_amdgcn_wmma_f32_16x16x4_f32(
                    false, a0, false, b, (short)0, acc0, false, false);
                acc1 = __builtin_amdgcn_wmma_f32_16x16x4_f32(
                    false, a1, false, b, (short)0, acc1, false, false);
            }
            const float bias = b_conv[nt * 16 + mrow] + b_skip[nt * 16 + mrow];
#pragma unroll
            for (int v = 0; v < 8; ++v) { acc0[v] += bias; acc1[v] += bias; }
        }

        // C/D layout: lane -> column nt*16+mrow; VGPR v -> row v + (lane>=16?8:0)
        const int mbase = (lane >> 4) * 8;
#pragma unroll
        for (int v = 0; v < 8; ++v) {
            const int g0 = row0 + mbase + v;
            const int g1 = g0 + 16;
            if (g0 < n_nodes) out[(size_t)g0 * RG_DIM + nt * 16 + mrow] = acc0[v];
            if (g1 < n_nodes) out[(size_t)g1 * RG_DIM + nt * 16 + mrow] = acc1[v];
        }
    }
}

// ---------------------------------------------------------------------------
// Edge gather/scatter: one wave32 per edge, 4 consecutive f32 per lane.
// Metadata scalarized via readfirstlane -> s_load; gather is one b128 per lane.
// hr (76.8 MB) + accumulator (25.6 MB) are L2-resident (192 MB L2).
// ---------------------------------------------------------------------------
__global__ __launch_bounds__(256)
void rgcn_edge_kernel(const int*   __restrict__ edge_index, // [2,E]
                      const int*   __restrict__ edge_type,  // [E]
                      const float* __restrict__ edge_weight,// [E]
                      const float* __restrict__ hr,         // [3,N,128]
                      float*       __restrict__ out,        // [N,128] (+=)
                      int n_edges, int n_nodes)
{
    const int t = blockIdx.x * blockDim.x + threadIdx.x;
    const int e = __builtin_amdgcn_readfirstlane(t >> 5);   // wave-uniform edge id
    if (e >= n_edges) return;
    const int lane = t & 31;

    const int   src = edge_index[e];
    const int   dst = edge_index[n_edges + e];
    const int   rel = edge_type[e];
    const float w   = edge_weight[e];

    const float4 m =
        ((const float4*)(hr + ((size_t)rel * n_nodes + src) * RG_DIM))[lane];

    float* q = out + (size_t)dst * RG_DIM + lane * 4;
    atomicAdd(q + 0, m.x * w);   // no-return global_atomic_add_f32
    atomicAdd(q + 1, m.y * w);
    atomicAdd(q + 2, m.z * w);
    atomicAdd(q + 3, m.w * w);
}

// ---------------------------------------------------------------------------
// Launch: weight pre-pack, then 2 x (fused GEMM, edge scatter).
// ws: hr [3*N*128] + h1 [N*128] + packed weights [10*PACK_M float2] ~ 103 MB.
// ---------------------------------------------------------------------------
extern "C" void kernel_launch(void* const* d_in, const int* in_sizes, int n_in,
                              void* d_out, int out_size, void* d_ws, size_t ws_size,
                              hipStream_t stream)
{
    const float* x           = (const float*)d_in[0];
    const int*   edge_index  = (const int*)  d_in[1];
    const int*   edge_type   = (const int*)  d_in[2];
    const float* edge_weight = (const float*)d_in[3];
    const float* w_rel       = (const float*)d_in[4]; // [2,3,128,128]
    const float* w_root      = (const float*)d_in[5]; // [2,128,128]
    const float* b_conv      = (const float*)d_in[6]; // [2,128]
    const float* w_skip      = (const float*)d_in[7]; // [2,128,128]
    const float* b_skip      = (const float*)d_in[8]; // [2,128]
    float*       out         = (float*)d_out;

    float*  hr    = (float*)d_ws;                          // [3,N,128]
    float*  h1    = hr + (size_t)3 * RG_NODES * RG_DIM;    // [N,128]
    float2* packw = (float2*)(h1 + (size_t)RG_NODES * RG_DIM); // [2][5][PACK_M]

    const int gemm_grid = (RG_NODES + 31) / 32;            // 1563
    const int edge_grid = (RG_EDGES * 32 + 255) / 256;     // 75000
    const int pack_grid = (10 * PACK_M + 255) / 256;       // 320

    rgcn_pack_w<<<pack_grid, 256, 0, stream>>>(w_rel, w_root, w_skip, packw);

    // ---- layer 0 (h = x, base -> h1) ----
    rgcn_gemm_kernel<<<gemm_grid, 128, 0, stream>>>(
        x, x, packw, b_conv, b_skip, hr, h1, RG_NODES);
    rgcn_edge_kernel<<<edge_grid, 256, 0, stream>>>(
        edge_index, edge_type, edge_weight, hr, h1, RG_EDGES, RG_NODES);

    // ---- layer 1 (h = h1, base -> d_out) ----
    rgcn_gemm_kernel<<<gemm_grid, 128, 0, stream>>>(
        h1, x, packw + (size_t)5 * PACK_M, b_conv + RG_DIM, b_skip + RG_DIM,
        hr, out, RG_NODES);
    rgcn_edge_kernel<<<edge_grid, 256, 0, stream>>>(
        edge_index, edge_type, edge_weight, hr, out, RG_EDGES, RG_NODES);
}